// GraphConvolution_26740466385631
// MI455X (gfx1250) — compile-verified
//
#include <hip/hip_runtime.h>
#include <hip/hip_bf16.h>

#define IN_F   128
#define OUT_F  64
#define LN_EPS 1e-5f
#define EDGES_PER_WAVE 4

typedef __attribute__((ext_vector_type(2))) float v2f;
typedef __attribute__((ext_vector_type(8))) float v8f;

// ---------------------------------------------------------------------------
// Kernel 1: build Wc[128][128] = [ weight (128x64) | res_w^T (128x64) ]
// ---------------------------------------------------------------------------
__global__ void gcn_prep_wc(const float* __restrict__ weight,
                            const float* __restrict__ res_w,
                            float* __restrict__ wc) {
    int gid = blockIdx.x * blockDim.x + threadIdx.x;   // 0 .. 16383
    if (gid < IN_F * 128) {
        int k = gid >> 7;        // 0..127
        int n = gid & 127;       // 0..127
        wc[gid] = (n < OUT_F) ? weight[k * OUT_F + n]
                              : res_w[(n - OUT_F) * IN_F + k];
    }
}

// ---------------------------------------------------------------------------
// Kernel 2: fused GEMM  C = x @ Wc   (C is [N,128]; left half -> support
// (+ zeroing of agg), right half -> d_out as residual + res_b).
// One wave per 16x16 tile, 4 waves / block sharing a 16x128 x-tile in LDS.
// grid = (ceil(N/16), 2), block = 128
// ---------------------------------------------------------------------------
__global__ void __launch_bounds__(128)
gcn_gemm_wmma(const float* __restrict__ x,
              const float* __restrict__ wc,
              const float* __restrict__ res_b,
              float* __restrict__ support,
              float* __restrict__ agg,
              float* __restrict__ out,
              int n_nodes) {
    __shared__ float lds_x[16][IN_F];   // 8 KB

    const int row0 = blockIdx.x * 16;
    const int lane = threadIdx.x & 31;
    const int wave = threadIdx.x >> 5;
    const int n0   = blockIdx.y * 64 + wave * 16;   // column tile start in Wc

    // Cooperative load of 16x128 x-tile: each of 128 threads loads 16 floats.
    {
        int r = threadIdx.x >> 3;          // 0..15
        int c = (threadIdx.x & 7) * 16;    // 0,16,...,112
        int row = row0 + r;
        if (row >= n_nodes) row = n_nodes - 1;   // clamp (no effect when N%16==0)
        const float4* src = (const float4*)(x + (size_t)row * IN_F + c);
        float4* dst = (float4*)(&lds_x[r][c]);
        dst[0] = src[0]; dst[1] = src[1]; dst[2] = src[2]; dst[3] = src[3];
    }
    __syncthreads();

    const int half = lane >> 4;     // 0: K+0..1, 1: K+2..3 (per WMMA f32 layout)
    const int l    = lane & 15;     // A: row M=l ; B: column N=n0+l

    v8f acc = {};
#pragma unroll 8
    for (int kk = 0; kk < IN_F; kk += 4) {
        const int ka = kk + half * 2;
        v2f a;
        a.x = lds_x[l][ka];
        a.y = lds_x[l][ka + 1];
        v2f b;
        b.x = wc[(size_t)ka * 128 + n0 + l];
        b.y = wc[(size_t)(ka + 1) * 128 + n0 + l];
        acc = __builtin_amdgcn_wmma_f32_16x16x4_f32(
            /*neg_a=*/false, a, /*neg_b=*/false, b,
            /*c_mod=*/(short)0, acc, /*reuse_a=*/false, /*reuse_b=*/false);
    }

    // C/D layout: VGPR r -> row (half*8 + r), column n0 + l
    const int col   = n0 + l;
    const int rbase = row0 + half * 8;

    if (row0 + 16 <= n_nodes) {
        // Fast path: whole tile in range, unguarded coalesced stores.
        if (blockIdx.y == 0) {
#pragma unroll
            for (int r = 0; r < 8; ++r) {
                size_t idx = (size_t)(rbase + r) * OUT_F + col;
                support[idx] = acc[r];
                agg[idx]     = 0.0f;          // fused accumulator zeroing
            }
        } else {
            const float rb = res_b[col - OUT_F];
#pragma unroll
            for (int r = 0; r < 8; ++r) {
                size_t idx = (size_t)(rbase + r) * OUT_F + (col - OUT_F);
                out[idx] = acc[r] + rb;
            }
        }
    } else {
        // Generic tail (unused when N % 16 == 0).
        if (blockIdx.y == 0) {
#pragma unroll
            for (int r = 0; r < 8; ++r) {
                int row = rbase + r;
                if (row < n_nodes) {
                    size_t idx = (size_t)row * OUT_F + col;
                    support[idx] = acc[r];
                    agg[idx]     = 0.0f;
                }
            }
        } else {
            const float rb = res_b[col - OUT_F];
#pragma unroll
            for (int r = 0; r < 8; ++r) {
                int row = rbase + r;
                if (row < n_nodes)
                    out[(size_t)row * OUT_F + (col - OUT_F)] = acc[r] + rb;
            }
        }
    }
}

// ---------------------------------------------------------------------------
// Kernel 3: SpMM scatter. One wave per EDGES_PER_WAVE edges; edge metadata
// forced uniform (SGPR) so it lowers to scalar loads; lanes carry 2 features.
// agg[dst][f] += w_e * support[src][f]
// ---------------------------------------------------------------------------
__global__ void gcn_spmm(const float* __restrict__ support,
                         const float* __restrict__ ew,
                         const int* __restrict__ esrc,
                         const int* __restrict__ edst,
                         float* __restrict__ agg,
                         int n_edges) {
    int wave_id = (int)(blockIdx.x * (blockDim.x >> 5)) + (int)(threadIdx.x >> 5);
    wave_id = __builtin_amdgcn_readfirstlane(wave_id);   // force SGPR (wave-uniform)
    const int lane = threadIdx.x & 31;
    const int f = lane * 2;
    const int e0 = wave_id * EDGES_PER_WAVE;

#pragma unroll
    for (int i = 0; i < EDGES_PER_WAVE; ++i) {
        int e = e0 + i;                       // uniform
        if (e >= n_edges) break;              // uniform branch
        int   s = esrc[e];                    // scalar loads (uniform address)
        int   d = edst[e];
        float w = ew[e];

        float2 sv = *(const float2*)(support + (size_t)s * OUT_F + f);
        float* dp = agg + (size_t)d * OUT_F + f;
        atomicAdd(dp,     w * sv.x);
        atomicAdd(dp + 1, w * sv.y);
    }
}

// ---------------------------------------------------------------------------
// Kernel 4: bias + LayerNorm + ReLU + residual. One wave per node.
// ---------------------------------------------------------------------------
__global__ void gcn_finalize(const float* __restrict__ agg,
                             const float* __restrict__ bias,
                             const float* __restrict__ gamma,
                             const float* __restrict__ beta,
                             float* __restrict__ out,
                             int n_nodes) {
    size_t gid = (size_t)blockIdx.x * blockDim.x + threadIdx.x;
    int node = (int)(gid >> 5);
    if (node >= n_nodes) return;
    int lane = (int)(gid & 31);
    int f = lane * 2;
    size_t base = (size_t)node * OUT_F + f;

    float2 v  = *(const float2*)(agg + base);
    float2 bb = *(const float2*)(bias + f);
    float2 gg = *(const float2*)(gamma + f);
    float2 be = *(const float2*)(beta + f);
    v.x += bb.x;
    v.y += bb.y;

    float s  = v.x + v.y;
    float sq = v.x * v.x + v.y * v.y;
#pragma unroll
    for (int off = 16; off > 0; off >>= 1) {
        s  += __shfl_xor(s,  off, 32);
        sq += __shfl_xor(sq, off, 32);
    }
    float mean = s * (1.0f / OUT_F);
    float var  = sq * (1.0f / OUT_F) - mean * mean;
    float inv  = rsqrtf(var + LN_EPS);

    float y0 = (v.x - mean) * inv * gg.x + be.x;
    float y1 = (v.y - mean) * inv * gg.y + be.y;
    y0 = fmaxf(y0, 0.0f);
    y1 = fmaxf(y1, 0.0f);

    float2 r = *(const float2*)(out + base);   // residual (incl. res_b) from GEMM
    out[base]     = y0 + r.x;
    out[base + 1] = y1 + r.y;
}

// ---------------------------------------------------------------------------
// Launch
// ---------------------------------------------------------------------------
extern "C" void kernel_launch(void* const* d_in, const int* in_sizes, int n_in,
                              void* d_out, int out_size, void* d_ws, size_t ws_size,
                              hipStream_t stream) {
    const float* x      = (const float*)d_in[0];
    const float* weight = (const float*)d_in[1];
    const float* bias   = (const float*)d_in[2];
    const float* gamma  = (const float*)d_in[3];
    const float* beta   = (const float*)d_in[4];
    const float* res_w  = (const float*)d_in[5];
    const float* res_b  = (const float*)d_in[6];
    const float* ew     = (const float*)d_in[7];
    const int*   esrc   = (const int*)d_in[8];
    const int*   edst   = (const int*)d_in[9];
    float*       out    = (float*)d_out;

    const int n_nodes = in_sizes[0] / IN_F;
    const int n_edges = in_sizes[7];

    // workspace layout: Wc (64 KB) | support (N*64 f32) | agg (N*64 f32)
    float* wc      = (float*)d_ws;
    float* support = wc + IN_F * 128;
    float* agg     = support + (size_t)n_nodes * OUT_F;

    gcn_prep_wc<<<(IN_F * 128 + 255) / 256, 256, 0, stream>>>(weight, res_w, wc);

    dim3 ggrid((n_nodes + 15) / 16, 2);
    gcn_gemm_wmma<<<ggrid, 128, 0, stream>>>(x, wc, res_b, support, agg, out, n_nodes);

    int waves_needed = (n_edges + EDGES_PER_WAVE - 1) / EDGES_PER_WAVE;
    int spmm_blocks  = (waves_needed + 7) / 8;          // 8 waves (256 thr) / block
    gcn_spmm<<<spmm_blocks, 256, 0, stream>>>(support, ew, esrc, edst, agg, n_edges);

    size_t fin_threads = (size_t)n_nodes * 32;
    gcn_finalize<<<(int)((fin_threads + 255) / 256), 256, 0, stream>>>(
        agg, bias, gamma, beta, out, n_nodes);
}